// SpikingNeuron_85899346048
// MI455X (gfx1250) — compile-verified
//
#include <hip/hip_runtime.h>
#include <stdint.h>
#include <stddef.h>

// LIF spiking-neuron scan, forward only.
//   fb = sigmoid(beta); m0 = init_rand * threshold
//   per t: m_new = fb*m + x[t]; spike = (m_new - thr >= 0); m = m_new - spike*thr
//   outputs: spike train and PRE-step membrane, both (N, T).
//
// Bandwidth-bound: 64 MB read + 128 MB write -> ~8.2 us floor at 23.3 TB/s.
// CDNA5 path: double-buffered async global->LDS DMA (ASYNCcnt pipelining with
// exact partial waits, since async loads retire in order) to coalesce the
// 1KB-strided (N,T) row reads; outputs go out as per-thread float4 stores and
// get line-merged in the 192 MB L2 (dirty working set ~16 MB).

#define T_STEPS   256
#define RPB       128            // rows per block == threads per block (4 wave32)
#define TCH       16             // timesteps per chunk
#define NCH       (T_STEPS / TCH)
#define STRIDE_F  18             // LDS row stride (floats): 8B-aligned, conflict-free
#define BUF_ELTS  (RPB * STRIDE_F)
#define LOADS_PER_THREAD 8       // RPB*TCH/2 b64 transfers / RPB threads

__global__ __launch_bounds__(RPB) void lif_scan_kernel(
    const float* __restrict__ x,          // (N, T)
    const float* __restrict__ beta_p,     // scalar
    const float* __restrict__ thr_p,      // scalar
    const float* __restrict__ init_rand,  // (N,)
    float* __restrict__ out_spk,          // (N, T)
    float* __restrict__ out_mem)          // (N, T)
{
    __shared__ float smem[2 * BUF_ELTS];  // 18 KB: two input tiles

    const int tid  = threadIdx.x;
    const int row0 = blockIdx.x * RPB;
    const int n    = row0 + tid;

    const float beta = beta_p[0];
    const float thr  = thr_p[0];
    const float fb   = 1.0f / (1.0f + __expf(-beta));   // sigmoid(beta)
    float m = init_rand[n] * thr;

    // Workgroup-relative LDS byte offset of the tile base.
    const uint32_t ldsBase = (uint32_t)(uintptr_t)(void*)smem;

    // Per-thread DMA descriptors, computed once: thread issues 8 b64 transfers
    // per chunk; lanes 0..7 cover one row's contiguous 64B segment (coalesced).
    uint64_t gbase[LOADS_PER_THREAD];
    uint32_t lbase[LOADS_PER_THREAD];
#pragma unroll
    for (int k = 0; k < LOADS_PER_THREAD; ++k) {
        int g    = k * RPB + tid;
        int row  = g >> 3;               // 8 x b64 per 16-float row chunk
        int pair = g & 7;
        gbase[k] = (uint64_t)(uintptr_t)(x + (size_t)(row0 + row) * T_STEPS + pair * 2);
        lbase[k] = ldsBase + (uint32_t)(row * STRIDE_F + pair * 2) * 4u;
    }

    // ---- prologue: stage chunk 0 into buffer 0 ----
#pragma unroll
    for (int k = 0; k < LOADS_PER_THREAD; ++k) {
        asm volatile("global_load_async_to_lds_b64 %0, %1, off"
                     :: "v"(lbase[k]), "v"(gbase[k]) : "memory");
    }

    float* outS = out_spk + (size_t)n * T_STEPS;
    float* outM = out_mem + (size_t)n * T_STEPS;

    for (int c = 0; c < NCH; ++c) {
        const int      cur      = c & 1;
        const uint32_t nxtOff   = (uint32_t)(((c + 1) & 1) * BUF_ELTS * 4);
        const uint64_t nxtDelta = (uint64_t)((c + 1) * TCH * 4);

        // ---- issue next chunk's DMA, then wait for ONLY the current chunk ----
        if (c + 1 < NCH) {
#pragma unroll
            for (int k = 0; k < LOADS_PER_THREAD; ++k) {
                asm volatile("global_load_async_to_lds_b64 %0, %1, off"
                             :: "v"(lbase[k] + nxtOff), "v"(gbase[k] + nxtDelta)
                             : "memory");
            }
            // Async loads retire in order: <=8 outstanding => chunk c landed.
            asm volatile("s_wait_asynccnt 8" ::: "memory");
        } else {
            asm volatile("s_wait_asynccnt 0" ::: "memory");
        }
        __syncthreads();   // every wave's chunk-c tile is visible in LDS

        // ---- serial LIF recurrence over this chunk ----
        const float* inT = smem + cur * BUF_ELTS + tid * STRIDE_F;
        const int    tc0 = c * TCH;
#pragma unroll
        for (int jj = 0; jj < TCH; jj += 4) {
            float sv[4], mv[4];
#pragma unroll
            for (int u = 0; u < 4; ++u) {
                float xv      = inT[jj + u];              // conflict-free (stride 18)
                float m_pre   = m;
                float m_new   = __builtin_fmaf(fb, m, xv);
                float surplus = m_new - thr;
                float spike   = (surplus >= 0.0f) ? 1.0f : 0.0f;
                m = __builtin_fmaf(-spike, thr, m_new);
                sv[u] = spike;
                mv[u] = m_pre;
            }
            // 16B-aligned per-thread stores; L2 merges the 8 segments per line.
            float4 s4 = make_float4(sv[0], sv[1], sv[2], sv[3]);
            float4 m4 = make_float4(mv[0], mv[1], mv[2], mv[3]);
            *reinterpret_cast<float4*>(outS + tc0 + jj) = s4;
            *reinterpret_cast<float4*>(outM + tc0 + jj) = m4;
        }
        __syncthreads();   // tile `cur` free before chunk c+2's DMA targets it
    }
}

extern "C" void kernel_launch(void* const* d_in, const int* in_sizes, int n_in,
                              void* d_out, int out_size, void* d_ws, size_t ws_size,
                              hipStream_t stream) {
    const float* x    = (const float*)d_in[0];   // (N, T) f32
    const float* beta = (const float*)d_in[1];   // scalar
    const float* thr  = (const float*)d_in[2];   // scalar
    const float* ir   = (const float*)d_in[3];   // (N,) f32

    const int N = in_sizes[3];                   // 65536
    float* out_spk = (float*)d_out;
    float* out_mem = out_spk + (size_t)N * T_STEPS;

    dim3 grid(N / RPB), block(RPB);
    hipLaunchKernelGGL(lif_scan_kernel, grid, block, 0, stream,
                       x, beta, thr, ir, out_spk, out_mem);
}